// NCHL_49151605736095
// MI455X (gfx1250) — compile-verified
//
#include <hip/hip_runtime.h>
#include <hip/hip_bf16.h>

typedef __attribute__((ext_vector_type(2))) float v2f;
typedef __attribute__((ext_vector_type(4))) float v4f;
typedef __attribute__((ext_vector_type(8))) float v8f;

#define NT 4            // 16-col tiles per wave -> wave computes 16x64
#define WAVES_PER_BLK 4 // 128 threads

static __device__ __forceinline__ v8f wmma_f32(v2f a, v2f b, v8f c) {
  return __builtin_amdgcn_wmma_f32_16x16x4_f32(
      /*neg_a=*/false, a, /*neg_b=*/false, b,
      /*c_mod=*/(short)0, c, /*reuse_a=*/false, /*reuse_b=*/false);
}

// Out[m,n] = tanh( sum_k X[m,k] * W[n,k] )   (i.e. tanh(X @ W^T))
// X: [M,K] row-major, W: [N,K] row-major, Out: [M,N] row-major.
// M,N multiples of 16; K multiple of 8.
//
// K is processed in chunks of 8 with a wave-consistent K-permutation:
// lane-half h loads floats [k0+4h, k0+4h+3] (one b128). WMMA #1 of the chunk
// consumes pair (.xy) = K-set {k0,k0+1,k0+4,k0+5}; WMMA #2 consumes (.zw) =
// K-set {k0+2,k0+3,k0+6,k0+7}. A and B use the same permutation, so the
// accumulated dot product over the chunk is exact.
__global__ __launch_bounds__(128) void nchl_gemm_tanh(const float* __restrict__ X,
                                                      const float* __restrict__ W,
                                                      float* __restrict__ Out,
                                                      int M, int N, int K) {
  const int lane = threadIdx.x & 31;
  const int wave = threadIdx.x >> 5;
  const int lid  = lane & 15;      // row (A) / col (B) within tile
  const int half = lane >> 4;
  const int m0 = blockIdx.y * 16;
  const int n0 = (blockIdx.x * WAVES_PER_BLK + wave) * (16 * NT);
  const int koff = half * 4;       // this lane's 4-float group within the chunk

  v8f acc[NT] = {};

  const float* xrow = X + (size_t)(m0 + lid) * K + koff;
  const float* wrow = W + (size_t)(n0 + lid) * K + koff;
  const size_t wstep = (size_t)16 * K;   // 16 rows per N-tile

  // --- software pipeline: preload chunk 0
  v4f a_c = *(const v4f*)(xrow);
  v4f b_c[NT];
#pragma unroll
  for (int t = 0; t < NT; ++t) b_c[t] = *(const v4f*)(wrow + (size_t)t * wstep);

  for (int k0 = 0; k0 + 8 < K; k0 += 8) {
    // issue next chunk's loads first so VMEM overlaps the WMMAs below
    v4f a_n = *(const v4f*)(xrow + k0 + 8);
    v4f b_n[NT];
#pragma unroll
    for (int t = 0; t < NT; ++t)
      b_n[t] = *(const v4f*)(wrow + (size_t)t * wstep + k0 + 8);

    const v2f alo = a_c.xy, ahi = a_c.zw;
#pragma unroll
    for (int t = 0; t < NT; ++t) {
      acc[t] = wmma_f32(alo, b_c[t].xy, acc[t]);
      acc[t] = wmma_f32(ahi, b_c[t].zw, acc[t]);
    }
    a_c = a_n;
#pragma unroll
    for (int t = 0; t < NT; ++t) b_c[t] = b_n[t];
  }

  // --- epilogue chunk
  {
    const v2f alo = a_c.xy, ahi = a_c.zw;
#pragma unroll
    for (int t = 0; t < NT; ++t) {
      acc[t] = wmma_f32(alo, b_c[t].xy, acc[t]);
      acc[t] = wmma_f32(ahi, b_c[t].zw, acc[t]);
    }
  }

  // D layout: VGPR r -> row m0 + r + 8*half, col n0 + 16t + lid
#pragma unroll
  for (int t = 0; t < NT; ++t) {
    const int col = n0 + 16 * t + lid;
#pragma unroll
    for (int r = 0; r < 8; ++r) {
      const int row = m0 + r + 8 * half;
      Out[(size_t)row * N + col] = tanhf(acc[t][r]);
    }
  }
}

// nW[j,i] = W[j,i] + eta_ij * (A_i + B_j + corr_ij + dec_ij)
// Vectorized: each thread handles 4 consecutive i. n_pre must be a multiple
// of 4*blockDim.x*gridDim.x partitioning (grids sized exactly below).
// The 4 hebb pre-rows span 20 contiguous floats (80B-aligned) -> five b128
// loads; W/nW move as b128.
__global__ __launch_bounds__(256) void nchl_hebb_update(const float* __restrict__ W,
                                                        const float* __restrict__ preP,
                                                        const float* __restrict__ postP,
                                                        const float* __restrict__ a_pre,
                                                        const float* __restrict__ a_post,
                                                        float* __restrict__ nW,
                                                        int n_pre) {
  const int j = blockIdx.y;                  // post neuron (row of W)
  const float p1 = postP[j * 5 + 1];
  const float p2 = postP[j * 5 + 2];
  const float p3 = postP[j * 5 + 3];
  const float p4 = postP[j * 5 + 4];
  const float apj = a_post[j];
  const float Bj = p1 * apj;
  const float Cj = (p2 == 1.0f) ? 1.0f : p2 * apj;
  const float Dj = p3;

  const int i0 = (blockIdx.x * blockDim.x + threadIdx.x) * 4;

  // 4 hebb rows = 20 contiguous floats starting at preP + 5*i0 (16B aligned)
  const v4f* pr = (const v4f*)(preP + (size_t)5 * i0);
  v4f pb0 = pr[0], pb1 = pr[1], pb2 = pr[2], pb3 = pr[3], pb4 = pr[4];
  float buf[20];
  *(v4f*)(buf + 0)  = pb0;
  *(v4f*)(buf + 4)  = pb1;
  *(v4f*)(buf + 8)  = pb2;
  *(v4f*)(buf + 12) = pb3;
  *(v4f*)(buf + 16) = pb4;

  const v4f ap = *(const v4f*)(a_pre + i0);
  const size_t idx = (size_t)j * n_pre + i0;
  const v4f w = *(const v4f*)(W + idx);

  v4f res;
#pragma unroll
  for (int r = 0; r < 4; ++r) {
    const float q0 = buf[5 * r + 0];
    const float q2 = buf[5 * r + 2];
    const float q3 = buf[5 * r + 3];
    const float q4 = buf[5 * r + 4];
    const float api = ap[r];
    const float Ai = q0 * api;
    const float Ci = (q2 == 1.0f) ? 1.0f : q2 * api;
    const float Di = q3;
    const float corr = (Ci == 1.0f && Cj == 1.0f) ? 0.0f : Ci * Cj;
    const float dec  = (Di == 1.0f && Dj == 1.0f) ? 0.0f : Di * Dj;
    const float dw = Ai + Bj + corr + dec;
    const float eta = 0.5f * (q4 + p4);
    res[r] = w[r] + eta * dw;
  }
  *(v4f*)(nW + idx) = res;
}

extern "C" void kernel_launch(void* const* d_in, const int* in_sizes, int n_in,
                              void* d_out, int out_size, void* d_ws, size_t ws_size,
                              hipStream_t stream) {
  (void)in_sizes; (void)n_in; (void)out_size; (void)ws_size;

  const float* x    = (const float*)d_in[0];  // [256,1024]
  const float* W1   = (const float*)d_in[1];  // [4096,1024]
  const float* W2   = (const float*)d_in[2];  // [4096,4096]
  const float* W3   = (const float*)d_in[3];  // [1024,4096]
  const float* hebb = (const float*)d_in[4];  // [10240,5]

  const int B = 256;
  const int N0 = 1024, N1 = 4096, N2 = 4096, N3 = 1024;

  float* h1 = (float*)d_ws;                    // [256,4096]
  float* h2 = h1 + (size_t)B * N1;             // [256,4096]

  float* out = (float*)d_out;
  float* y   = out;                            // [256,1024]
  float* nW1 = y + (size_t)B * N3;             // [4096,1024]
  float* nW2 = nW1 + (size_t)N1 * N0;          // [4096,4096]
  float* nW3 = nW2 + (size_t)N2 * N1;          // [1024,4096]

  const float* P0 = hebb;                      // rows [0,1024)
  const float* P1 = hebb + (size_t)1024 * 5;   // rows [1024,5120)
  const float* P2 = hebb + (size_t)5120 * 5;   // rows [5120,9216)
  const float* P3 = hebb + (size_t)9216 * 5;   // rows [9216,10240)

  // ---- Forward: h1 = tanh(x @ W1^T), h2 = tanh(h1 @ W2^T), y = tanh(h2 @ W3^T)
  {
    dim3 blk(128);
    dim3 g1(N1 / (16 * NT * WAVES_PER_BLK), B / 16);  // 16 x 16
    nchl_gemm_tanh<<<g1, blk, 0, stream>>>(x, W1, h1, B, N1, N0);
    dim3 g2(N2 / (16 * NT * WAVES_PER_BLK), B / 16);  // 16 x 16
    nchl_gemm_tanh<<<g2, blk, 0, stream>>>(h1, W2, h2, B, N2, N1);
    dim3 g3(N3 / (16 * NT * WAVES_PER_BLK), B / 16);  // 4 x 16
    nchl_gemm_tanh<<<g3, blk, 0, stream>>>(h2, W3, y, B, N3, N2);
  }

  // ---- Hebbian updates (activations = row 0 of x, h1, h2, y)
  // Each thread covers 4 pre-neurons: gridDim.x = n_pre / (256*4).
  {
    dim3 blk(256);
    nchl_hebb_update<<<dim3(N0 / 1024, N1), blk, 0, stream>>>(W1, P0, P1, x, h1, nW1, N0);
    nchl_hebb_update<<<dim3(N1 / 1024, N2), blk, 0, stream>>>(W2, P1, P2, h1, h2, nW2, N1);
    nchl_hebb_update<<<dim3(N2 / 1024, N3), blk, 0, stream>>>(W3, P2, P3, h2, y, nW3, N2);
  }
}